// S4Layer_33260226740395
// MI455X (gfx1250) — compile-verified
//
#include <hip/hip_runtime.h>
#include <hip/hip_bf16.h>

// ---------------------------------------------------------------------------
// S4 layer on MI455X (gfx1250), f32 end-to-end via V_WMMA_F32_16X16X4_F32.
//
// Pipeline (all on `stream`, implicit ordering between kernels):
//   prep   : A_bar = I + DT*A, powers A_bar^0..64 (WMMA power iteration),
//            stacked-powers matrix P (column-major), B_bar = DT*B
//   ub     : uB[t,b,n]  = u[b,t,:] . B_bar[n,:]          (WMMA GEMM)
//   gagg   : g_c        = F_chunk  . P                   (WMMA GEMM, no chain)
//   bscan  : 64-step boundary scan with A^64             (WMMA, 1 WG, LDS)
//   states : per-chunk 64-step state reconstruction      (WMMA, 64 WGs par.)
//   yout   : y = S.C^T + u.D^T                           (WMMA GEMM)
// ---------------------------------------------------------------------------

typedef __attribute__((ext_vector_type(2))) float v2f;
typedef __attribute__((ext_vector_type(8))) float v8f;

#define S4_DT 0.001f
#define TT 4096      // time steps
#define NB 16        // batch
#define NS 64        // state dim
#define NI 256       // input dim
#define NO 256       // output dim
#define LCH 64       // chunk length
#define NCH 64       // number of chunks (TT / LCH)

__device__ __forceinline__ v8f wmma4(v2f a, v2f b, v8f c) {
  // D = A(16x4,f32) * B(4x16,f32) + C(16x16,f32)
  return __builtin_amdgcn_wmma_f32_16x16x4_f32(
      /*neg_a=*/false, a, /*neg_b=*/false, b,
      /*c_mod=*/(short)0, c, /*reuse_a=*/false, /*reuse_b=*/false);
}

__device__ __forceinline__ v2f load2(const float* p) { return *(const v2f*)p; }

// C/D 16x16 f32 tile layout: VGPR v -> row (v + 8*half), col (n0 + l16)
__device__ __forceinline__ void store_c(float* dst, int rs, int n0, v8f acc,
                                        int l16, int half) {
#pragma unroll
  for (int v = 0; v < 8; ++v)
    dst[(v + 8 * half) * rs + n0 + l16] = acc[v];
}

__device__ __forceinline__ v8f load_c(const float* src, int rs, int n0,
                                      int l16, int half) {
  v8f r;
#pragma unroll
  for (int v = 0; v < 8; ++v)
    r[v] = src[(v + 8 * half) * rs + n0 + l16];
  return r;
}

// ---------------------------------------------------------------------------
// prep: A_bar, its powers, stacked-powers matrix, scaled B.
//   Apow  : 65 matrices of 64x64, row-major
//   Pcat  : column-major [n][kappa], kappa = j*64+m, value = A^{63-j}[n][m]
//   Bsc   : DT * B (64x256 row-major)
// One workgroup, 16 waves; wave w owns 16x16 tile (w>>2, w&3).
// ---------------------------------------------------------------------------
__global__ __launch_bounds__(512) void s4_prep_kernel(
    const float* __restrict__ A, const float* __restrict__ B,
    float* __restrict__ Abar, float* __restrict__ Apow,
    float* __restrict__ Pcat, float* __restrict__ Bsc) {
  __shared__ float sPrev[2][NS * NS];  // power double-buffer
  __shared__ float sAbT[NS * NS];      // A_bar column-major (B-operand layout)

  const int tid = threadIdx.x;
  for (int idx = tid; idx < NS * NS; idx += 512) {
    int r = idx >> 6, c = idx & 63;
    float v = A[idx] * S4_DT + (r == c ? 1.0f : 0.0f);
    Abar[idx] = v;
    sPrev[0][idx] = v;
    sAbT[c * NS + r] = v;
    Apow[0 * NS * NS + idx] = (r == c) ? 1.0f : 0.0f;
    Apow[1 * NS * NS + idx] = v;
  }
  for (int idx = tid; idx < NS * NI; idx += 512) Bsc[idx] = B[idx] * S4_DT;
  __syncthreads();

  const int wave = tid >> 5, lane = tid & 31;
  const int l16 = lane & 15, half = lane >> 4;
  const int m0 = (wave >> 2) * 16, n0 = (wave & 3) * 16;

  int cur = 0;
  for (int k = 2; k <= LCH; ++k) {  // A^k = A^{k-1} * A_bar
    v8f acc = {};
#pragma unroll
    for (int kk = 0; kk < 16; ++kk) {
      v2f a = load2(&sPrev[cur][(m0 + l16) * NS + 4 * kk + 2 * half]);
      v2f b = load2(&sAbT[(n0 + l16) * NS + 4 * kk + 2 * half]);
      acc = wmma4(a, b, acc);
    }
    int nxt = cur ^ 1;
#pragma unroll
    for (int v = 0; v < 8; ++v) {
      int row = m0 + v + 8 * half;
      float val = acc[v];
      sPrev[nxt][row * NS + n0 + l16] = val;
      Apow[k * NS * NS + row * NS + n0 + l16] = val;
    }
    __syncthreads();
    cur = nxt;
  }

  // Pcat[n*4096 + j*64 + m] = A^{63-j}[n][m]  (column-major over kappa)
  for (int idx = tid; idx < NS * (LCH * NS); idx += 512) {
    int n = idx >> 12, rest = idx & 4095, j = rest >> 6, m = rest & 63;
    Pcat[idx] = Apow[(63 - j) * NS * NS + n * NS + m];
  }
}

// ---------------------------------------------------------------------------
// ub: uB[t,b,n] = sum_i u[b,t,i] * Bsc[n,i].  One wave per t (16x64 tile).
// ---------------------------------------------------------------------------
__global__ __launch_bounds__(128) void s4_ub_kernel(
    const float* __restrict__ u, const float* __restrict__ Bsc,
    float* __restrict__ uB) {
  const int wave = threadIdx.x >> 5, lane = threadIdx.x & 31;
  const int l16 = lane & 15, half = lane >> 4;
  const int t = blockIdx.x * 4 + wave;

  const float* urow = u + (size_t)l16 * TT * NI + (size_t)t * NI;  // b = l16
  v8f acc[4] = {{}, {}, {}, {}};
  for (int kk = 0; kk < NI / 4; ++kk) {
    v2f a = load2(urow + 4 * kk + 2 * half);
#pragma unroll
    for (int j = 0; j < 4; ++j) {
      v2f b = load2(Bsc + (j * 16 + l16) * NI + 4 * kk + 2 * half);
      acc[j] = wmma4(a, b, acc[j]);
    }
  }
  float* dst = uB + (size_t)t * NB * NS;
#pragma unroll
  for (int j = 0; j < 4; ++j) store_c(dst, NS, j * 16, acc[j], l16, half);
}

// ---------------------------------------------------------------------------
// gagg: g_c[b,n] = sum_{j,m} uB[t0+j,b,m] * A^{63-j}[n,m]
//       = (16 x 4096) @ Pcat  — fully parallel, no serial chain.
// One wave per chunk.
// ---------------------------------------------------------------------------
__global__ __launch_bounds__(128) void s4_gagg_kernel(
    const float* __restrict__ uB, const float* __restrict__ Pcat,
    float* __restrict__ G) {
  const int wave = threadIdx.x >> 5, lane = threadIdx.x & 31;
  const int l16 = lane & 15, half = lane >> 4;
  const int c = blockIdx.x * 4 + wave;

  const float* F = uB + (size_t)c * LCH * NB * NS;  // [j][b][m]
  v8f acc[4] = {{}, {}, {}, {}};
  for (int kk = 0; kk < (LCH * NS) / 4; ++kk) {
    int kap = 4 * kk + 2 * half;
    int j = kap >> 6, m = kap & 63;
    v2f a = load2(F + j * NB * NS + l16 * NS + m);
#pragma unroll
    for (int jn = 0; jn < 4; ++jn) {
      v2f b = load2(Pcat + (size_t)(jn * 16 + l16) * (LCH * NS) + kap);
      acc[jn] = wmma4(a, b, acc[jn]);
    }
  }
  float* dst = G + (size_t)c * NB * NS;
#pragma unroll
  for (int j = 0; j < 4; ++j) store_c(dst, NS, j * 16, acc[j], l16, half);
}

// ---------------------------------------------------------------------------
// bscan: 64-step boundary scan  Sb_{c+1} = Sb_c @ (A^64)^T + G_c.
// Stores chunk-start states and x_final. One workgroup, 4 waves, LDS dbuf.
// ---------------------------------------------------------------------------
__global__ __launch_bounds__(128) void s4_bscan_kernel(
    const float* __restrict__ x0, const float* __restrict__ G,
    const float* __restrict__ Apow, float* __restrict__ Sbound,
    float* __restrict__ xfinal) {
  __shared__ float sb[2][NB * NS];
  const int tid = threadIdx.x;
  const int wave = tid >> 5, lane = tid & 31;
  const int l16 = lane & 15, half = lane >> 4;
  const int n0 = wave * 16;
  const float* AL = Apow + (size_t)LCH * NS * NS;  // A^64 row-major = BT layout

  for (int idx = tid; idx < NB * NS; idx += 128) sb[0][idx] = x0[idx];
  __syncthreads();

  int cur = 0;
  for (int c = 0; c < NCH; ++c) {
    for (int idx = tid; idx < NB * NS; idx += 128)
      Sbound[(size_t)c * NB * NS + idx] = sb[cur][idx];
    v8f acc = load_c(G + (size_t)c * NB * NS, NS, n0, l16, half);
#pragma unroll
    for (int kk = 0; kk < 16; ++kk) {
      v2f a = load2(&sb[cur][l16 * NS + 4 * kk + 2 * half]);
      v2f b = load2(AL + (n0 + l16) * NS + 4 * kk + 2 * half);
      acc = wmma4(a, b, acc);
    }
    int nxt = cur ^ 1;
#pragma unroll
    for (int v = 0; v < 8; ++v)
      sb[nxt][(v + 8 * half) * NS + n0 + l16] = acc[v];
    __syncthreads();
    cur = nxt;
  }
  for (int idx = tid; idx < NB * NS; idx += 128) xfinal[idx] = sb[cur][idx];
}

// ---------------------------------------------------------------------------
// states: within-chunk reconstruction, 64 chunks in parallel.
//   s_{t0}   = Sbound[c];  for k: store s -> S[t0+k];  s = s @ Abar^T + uB[t0+k]
// One workgroup per chunk (4 waves share state via LDS).
// ---------------------------------------------------------------------------
__global__ __launch_bounds__(128) void s4_states_kernel(
    const float* __restrict__ uB, const float* __restrict__ Sbound,
    const float* __restrict__ Abar, float* __restrict__ S) {
  __shared__ float sb[2][NB * NS];
  const int tid = threadIdx.x;
  const int wave = tid >> 5, lane = tid & 31;
  const int l16 = lane & 15, half = lane >> 4;
  const int n0 = wave * 16;
  const int c = blockIdx.x;
  const int t0 = c * LCH;

  for (int idx = tid; idx < NB * NS; idx += 128)
    sb[0][idx] = Sbound[(size_t)c * NB * NS + idx];
  __syncthreads();

  int cur = 0;
  for (int k = 0; k < LCH; ++k) {
    float* dst = S + (size_t)(t0 + k) * NB * NS;
    for (int idx = tid; idx < NB * NS; idx += 128) dst[idx] = sb[cur][idx];
    // acc starts as the forcing term uB[t0+k] (C-layout gather)
    v8f acc = load_c(uB + (size_t)(t0 + k) * NB * NS, NS, n0, l16, half);
#pragma unroll
    for (int kk = 0; kk < 16; ++kk) {
      v2f a = load2(&sb[cur][l16 * NS + 4 * kk + 2 * half]);
      v2f b = load2(Abar + (n0 + l16) * NS + 4 * kk + 2 * half);  // Abar[n][k]
      acc = wmma4(a, b, acc);
    }
    int nxt = cur ^ 1;
#pragma unroll
    for (int v = 0; v < 8; ++v)
      sb[nxt][(v + 8 * half) * NS + n0 + l16] = acc[v];
    __syncthreads();
    cur = nxt;
  }
}

// ---------------------------------------------------------------------------
// yout: y[b,t,o] = S[t,b,:] . C[o,:]  +  u[b,t,:] . D[o,:]
// One wave per (t, 64-wide o-group); 4 o-tiles share A-operand loads.
// ---------------------------------------------------------------------------
__global__ __launch_bounds__(128) void s4_y_kernel(
    const float* __restrict__ S, const float* __restrict__ u,
    const float* __restrict__ C, const float* __restrict__ D,
    float* __restrict__ y) {
  const int wave = threadIdx.x >> 5, lane = threadIdx.x & 31;
  const int l16 = lane & 15, half = lane >> 4;
  const int t = blockIdx.x;
  const int o0 = wave * 64;

  v8f acc[4] = {{}, {}, {}, {}};

  const float* srow = S + (size_t)t * NB * NS + l16 * NS;  // b = l16
#pragma unroll
  for (int kk = 0; kk < NS / 4; ++kk) {
    v2f a = load2(srow + 4 * kk + 2 * half);
#pragma unroll
    for (int j = 0; j < 4; ++j) {
      v2f b = load2(C + (o0 + 16 * j + l16) * NS + 4 * kk + 2 * half);
      acc[j] = wmma4(a, b, acc[j]);
    }
  }

  const float* urow = u + (size_t)l16 * TT * NI + (size_t)t * NI;
  for (int kk = 0; kk < NI / 4; ++kk) {
    v2f a = load2(urow + 4 * kk + 2 * half);
#pragma unroll
    for (int j = 0; j < 4; ++j) {
      v2f b = load2(D + (o0 + 16 * j + l16) * NI + 4 * kk + 2 * half);
      acc[j] = wmma4(a, b, acc[j]);
    }
  }

#pragma unroll
  for (int j = 0; j < 4; ++j)
#pragma unroll
    for (int v = 0; v < 8; ++v)
      y[(size_t)(v + 8 * half) * TT * NO + (size_t)t * NO + o0 + 16 * j + l16] =
          acc[j][v];
}

// ---------------------------------------------------------------------------
extern "C" void kernel_launch(void* const* d_in, const int* in_sizes, int n_in,
                              void* d_out, int out_size, void* d_ws,
                              size_t ws_size, hipStream_t stream) {
  const float* u  = (const float*)d_in[0];  // (16, 4096, 256)
  const float* x0 = (const float*)d_in[1];  // (16, 64)
  const float* A  = (const float*)d_in[2];  // (64, 64)
  const float* B  = (const float*)d_in[3];  // (64, 256)
  const float* C  = (const float*)d_in[4];  // (256, 64)
  const float* D  = (const float*)d_in[5];  // (256, 256)

  float* y      = (float*)d_out;                        // (16, 4096, 256)
  float* xfinal = (float*)d_out + (size_t)NB * TT * NO; // (16, 64)

  // Workspace layout (floats); total ~34.6 MB
  float* ws = (float*)d_ws;
  float* Abar   = ws;                 ws += NS * NS;            //  16 KB
  float* Apow   = ws;                 ws += (LCH + 1) * NS * NS;// 1.06 MB
  float* Pcat   = ws;                 ws += NS * LCH * NS;      //   1 MB
  float* Bsc    = ws;                 ws += NS * NI;            //  64 KB
  float* uB     = ws;                 ws += (size_t)TT * NB * NS; // 16.8 MB
  float* Sst    = ws;                 ws += (size_t)TT * NB * NS; // 16.8 MB
  float* G      = ws;                 ws += NCH * NB * NS;      // 256 KB
  float* Sbound = ws;                 ws += NCH * NB * NS;      // 256 KB

  s4_prep_kernel<<<1, 512, 0, stream>>>(A, B, Abar, Apow, Pcat, Bsc);
  s4_ub_kernel<<<TT / 4, 128, 0, stream>>>(u, Bsc, uB);
  s4_gagg_kernel<<<NCH / 4, 128, 0, stream>>>(uB, Pcat, G);
  s4_bscan_kernel<<<1, 128, 0, stream>>>(x0, G, Apow, Sbound, xfinal);
  s4_states_kernel<<<NCH, 128, 0, stream>>>(uB, Sbound, Abar, Sst);
  s4_y_kernel<<<TT, 128, 0, stream>>>(Sst, u, C, D, y);
}